// GATv2Attention_64656437674325
// MI455X (gfx1250) — compile-verified
//
#include <hip/hip_runtime.h>
#include <hip/hip_bf16.h>

#define N_NODES 1024
#define EMBED   256
#define NHEAD   8
#define NHID    32
#define SLOPE   0.2f
#define NEGBIG  -1e30f

typedef float v2f __attribute__((ext_vector_type(2)));
typedef float v4f __attribute__((ext_vector_type(4)));
typedef float v8f __attribute__((ext_vector_type(8)));
typedef __attribute__((address_space(3))) float lds_f;
typedef __attribute__((address_space(3))) v4f   lds_v4f;

// LDS tile: 32 j-rows x 32 d-cols, row stride 36 words (144 B, 16B-aligned so
// both async B128 writes and ds_load_b128 reads are legal), double-buffered,
// private per wave (= per head).
#define TSTR   36
#define TWORDS (32 * TSTR)   // 1152 words per buffer

// ---------------------------------------------------------------------------
// Kernel 0: clear dense adjacency bitmask (1024*1024 bits = 32768 u32)
// ---------------------------------------------------------------------------
__global__ void gat_adj_clear(unsigned* __restrict__ adj) {
    adj[blockIdx.x * 256 + threadIdx.x] = 0u;
}

// ---------------------------------------------------------------------------
// Kernel 1: scatter edges (r = dest row, c = source col) into bitmask
// ---------------------------------------------------------------------------
__global__ void gat_adj_scatter(const int* __restrict__ ei,
                                unsigned* __restrict__ adj, int E) {
    int e = blockIdx.x * blockDim.x + threadIdx.x;
    if (e < E) {
        int r = ei[2 * e + 0];
        int c = ei[2 * e + 1];
        atomicOr(&adj[r * (N_NODES / 32) + (c >> 5)], 1u << (c & 31));
    }
}

// ---------------------------------------------------------------------------
// Kernel 2: projections g = h @ W via native fp32 WMMA (V_WMMA_F32_16X16X4_F32)
// One wave32 per 16x16 output tile, K swept in steps of 4. Exact fp32.
// ---------------------------------------------------------------------------
__global__ void gat_proj_wmma(const float* __restrict__ h,
                              const float* __restrict__ Wl,
                              const float* __restrict__ Wr,
                              float* __restrict__ gl,
                              float* __restrict__ gr) {
    const float* W = (blockIdx.z == 0) ? Wl : Wr;
    float*       g = (blockIdx.z == 0) ? gl : gr;

    const int n0   = blockIdx.x * 16;
    const int m0   = blockIdx.y * 16;
    const int lane = threadIdx.x & 31;
    const int hi   = lane >> 4;      // 0: K pair {0,1}, 1: K pair {2,3}
    const int lo   = lane & 15;      // A: row M ; B: col N

    v8f c = {};
    const float* hrow = h + (m0 + lo) * EMBED;

    #pragma unroll 8
    for (int k = 0; k < EMBED; k += 4) {
        const int kb = k + (hi << 1);
        v2f a, b;
        a.x = hrow[kb + 0];
        a.y = hrow[kb + 1];
        b.x = W[(kb + 0) * EMBED + n0 + lo];
        b.y = W[(kb + 1) * EMBED + n0 + lo];
        c = __builtin_amdgcn_wmma_f32_16x16x4_f32(
                false, a, false, b, (short)0, c, false, false);
    }

    #pragma unroll
    for (int v = 0; v < 8; ++v) {
        g[(m0 + v + 8 * hi) * EMBED + n0 + lo] = c[v];
    }
}

// ---------------------------------------------------------------------------
// Kernel 3: fused scores + mask + online softmax + aggregation.
// One block (8 waves) per destination row i; wave w == head h.
// Per 32-j tile:
//   * g_l sub-tile staged with 8x global_load_async_to_lds_b128 (ASYNCcnt)
//     into a per-wave LDS double-buffer; next tile prefetched during compute.
//   * score phase : lane == j, row pulled as 8x ds_load_b128, serial d walk
//   * softmax     : one tile-max + one tile-sum butterfly per 32 j
//   * aggregation : lane == d, 32 readlane broadcasts of p, LDS row FMAs
// ---------------------------------------------------------------------------
__global__ void gat_attn_fused(const float* __restrict__ gl,
                               const float* __restrict__ gr,
                               const float* __restrict__ attn_w,
                               const unsigned* __restrict__ adj,
                               float* __restrict__ out) {
    __shared__ float smem[NHEAD * 2 * TWORDS];   // 73.7 KB of the 320 KB WGP LDS

    const int i    = blockIdx.x;
    const int tid  = threadIdx.x;
    const int lane = tid & 31;
    const int w    = __builtin_amdgcn_readfirstlane(tid >> 5);  // head (scalar)

    lds_f* sbase   = (lds_f*)smem;                // raw LDS offset pointer
    lds_f* wavebuf = sbase + w * 2 * TWORDS;      // this wave's private region

    // wave-uniform vectors -> SGPRs via SMEM
    float grd[NHID], wd[NHID];
    #pragma unroll
    for (int d = 0; d < NHID; ++d) {
        grd[d] = gr[i * EMBED + (w << 5) + d];
        wd[d]  = attn_w[d];
    }

    // async-stage one 32x32 g_l sub-tile (head w's columns) into buffer `buf`
    // 8 B128 ops: instruction q moves rows 4q..4q+3 (lane -> row 4q+(l>>3),
    // 4-float group (l&7)*4); 16B-aligned on both global and LDS sides.
    const int rrow = lane >> 3;          // 0..3  row within quad
    const int rcol = (lane & 7) << 2;    // 0,4,...,28 column group
    auto issue_tile = [&](int t, int buf) {
        const float* src = gl + ((t << 5) + rrow) * EMBED + (w << 5) + rcol;
        lds_f*       dst = wavebuf + buf * TWORDS + rrow * TSTR + rcol;
        #pragma unroll
        for (int q = 0; q < 8; ++q) {
            asm volatile("global_load_async_to_lds_b128 %0, %1, off"
                         :: "v"(dst + q * 4 * TSTR),
                            "v"((unsigned long long)(src + q * 4 * EMBED))
                         : "memory");
        }
    };

    float m   = -3.0e38f;
    float l   = 0.0f;
    float acc = 0.0f;

    issue_tile(0, 0);

    const unsigned* arow = adj + i * (N_NODES / 32);

    for (int t = 0; t < N_NODES / 32; ++t) {
        const int buf = t & 1;
        asm volatile("s_wait_asynccnt 0x0" ::: "memory");
        if (t + 1 < N_NODES / 32) issue_tile(t + 1, buf ^ 1);

        lds_f* T = wavebuf + buf * TWORDS;
        const unsigned mask = arow[t];                 // uniform -> s_load

        // ---- scores: lane plays j, row read as 8 x b128, serial d walk ----
        float ssum = 0.0f;
        #pragma unroll
        for (int k = 0; k < NHID / 4; ++k) {
            v4f q = *(lds_v4f*)(T + lane * TSTR + 4 * k);
            #pragma unroll
            for (int u = 0; u < 4; ++u) {
                const int d  = 4 * k + u;
                float x  = q[u] + grd[d];
                float lr = fmaxf(x, SLOPE * x);        // leaky_relu, slope<1
                ssum = fmaf(wd[d], lr, ssum);
            }
        }
        float e = ((mask >> lane) & 1u) ? ssum : NEGBIG;

        // ---- tile max + online softmax state ----
        float tm = e;
        #pragma unroll
        for (int off = 16; off >= 1; off >>= 1)
            tm = fmaxf(tm, __shfl_xor(tm, off, 32));
        const float mn    = fmaxf(m, tm);
        const float scale = __expf(m - mn);
        const float p     = __expf(e - mn);

        float ps = p;
        #pragma unroll
        for (int off = 16; off >= 1; off >>= 1)
            ps += __shfl_xor(ps, off, 32);
        l = l * scale + ps;

        // ---- aggregation: lane plays d (conflict-free row reads) ----
        acc *= scale;
        #pragma unroll
        for (int jj = 0; jj < 32; ++jj) {
            const float pj = __shfl(p, jj, 32);        // readlane broadcast
            acc = fmaf(pj, T[jj * TSTR + lane], acc);
        }
        m = mn;
    }

    out[i * EMBED + (w << 5) + lane] = acc / l;
}

// ---------------------------------------------------------------------------
// Launch
// ---------------------------------------------------------------------------
extern "C" void kernel_launch(void* const* d_in, const int* in_sizes, int n_in,
                              void* d_out, int out_size, void* d_ws, size_t ws_size,
                              hipStream_t stream) {
    const float* h      = (const float*)d_in[0];
    const int*   ei     = (const int*)  d_in[1];
    const float* Wl     = (const float*)d_in[2];
    const float* Wr     = (const float*)d_in[3];
    const float* attn_w = (const float*)d_in[4];
    float*       out    = (float*)d_out;

    const int E = in_sizes[1] / 2;

    // workspace: g_l (1 MB) | g_r (1 MB) | adjacency bitmask (128 KB)
    float*    gl  = (float*)d_ws;
    float*    gr  = gl + N_NODES * EMBED;
    unsigned* adj = (unsigned*)(gr + N_NODES * EMBED);

    gat_adj_clear<<<dim3((N_NODES * N_NODES / 32) / 256), dim3(256), 0, stream>>>(adj);
    gat_adj_scatter<<<dim3((E + 255) / 256), dim3(256), 0, stream>>>(ei, adj, E);
    gat_proj_wmma<<<dim3(EMBED / 16, N_NODES / 16, 2), dim3(32), 0, stream>>>(
        h, Wl, Wr, gl, gr);
    gat_attn_fused<<<dim3(N_NODES), dim3(256), 0, stream>>>(gl, gr, attn_w, adj, out);
}